// SpatialGuidedConditionCluster_86414741995787
// MI455X (gfx1250) — compile-verified
//
#include <hip/hip_runtime.h>
#include <hip/hip_bf16.h>
#include <math.h>

typedef float v2f __attribute__((ext_vector_type(2)));
typedef float v8f __attribute__((ext_vector_type(8)));

#define WMMA_F32_16x16x4(a, b, c) \
  __builtin_amdgcn_wmma_f32_16x16x4_f32(false, (a), false, (b), (short)0, (c), false, false)

// ---------------------------------------------------------------------------
// Geometry constants
// ---------------------------------------------------------------------------
#define B0   2
#define CX   64
#define S    40
#define NSP  64000          // 40*40*40
#define EC   128            // heads*head_dim
#define HEADS 4
#define HD   32             // head dim
#define OUTC 256
#define M8   8              // 2*2*2 pool centers
#define NBP  (B0*HEADS)     // 8 "batch*head" groups

// ---------------------------------------------------------------------------
// 0) zero the num/den accumulators (harness does not re-zero ws between runs)
// ---------------------------------------------------------------------------
__global__ void __launch_bounds__(256) zero_kernel(float* __restrict__ num,
                                                   float* __restrict__ den) {
  int t = threadIdx.x;
  for (int i = t; i < NBP * M8 * HD; i += 256) num[i] = 0.0f;
  if (t < NBP * M8) den[t] = 0.0f;
}

// ---------------------------------------------------------------------------
// 1) pool raw inputs: bar[b][i][m] = mean over 20x20x20 region (m = wb*4+hb*2+db,
//    with w = 2nd spatial axis due to the h<->w swap in split_heads)
// ---------------------------------------------------------------------------
__global__ void __launch_bounds__(256) pool_kernel(const float* __restrict__ x,
                                                   const float* __restrict__ cond,
                                                   float* __restrict__ xbar,
                                                   float* __restrict__ cbar) {
  const int i = blockIdx.x;            // channel 0..63
  const int b = blockIdx.y;            // batch 0..1
  const float* src = (blockIdx.z == 0 ? x : cond) + (size_t)(b * CX + i) * NSP;
  float* dst = (blockIdx.z == 0 ? xbar : cbar) + (b * CX + i) * M8;
  __shared__ float red[256];
  const int tid = threadIdx.x;
  for (int m = 0; m < M8; ++m) {
    const int wb = (m >> 2) & 1, hb = (m >> 1) & 1, db = m & 1;
    float acc = 0.0f;
    for (int idx = tid; idx < 8000; idx += 256) {
      int hi = idx / 400, wi = (idx / 20) % 20, di = idx % 20;
      int h = hb * 20 + hi, w = wb * 20 + wi, d = db * 20 + di;
      acc += src[(h * S + w) * S + d];
    }
    red[tid] = acc;
    __syncthreads();
    for (int s = 128; s > 0; s >>= 1) {
      if (tid < s) red[tid] += red[tid + s];
      __syncthreads();
    }
    if (tid == 0) dst[m] = red[0] * (1.0f / 8000.0f);
    __syncthreads();
  }
}

// ---------------------------------------------------------------------------
// 2) centers: push pooled means through the (linear) encoders.
//    cXn = l2-normalized x-centers, cC = raw cond-centers. grid=64, block=32.
// ---------------------------------------------------------------------------
__global__ void __launch_bounds__(32) centers_kernel(
    const float* __restrict__ xbar, const float* __restrict__ cbar,
    const float* __restrict__ xw, const float* __restrict__ xb,
    const float* __restrict__ cw, const float* __restrict__ cb,
    float* __restrict__ cXn, float* __restrict__ cC) {
  const int g = blockIdx.x;                 // (bp, m)
  const int bp = g >> 3, m = g & 7;
  const int b = bp >> 2, e = bp & 3;
  const int c = threadIdx.x;
  const int o = e * HD + c;
  float cx = xb[o], cc = cb[o];
  for (int i = 0; i < CX; ++i) {
    float px = xbar[(b * CX + i) * M8 + m];
    float pc = cbar[(b * CX + i) * M8 + m];
    cx += xw[o * CX + i] * px;
    cc += cw[o * CX + i] * pc;
  }
  __shared__ float red[32];
  red[c] = cx * cx;
  __syncthreads();
  for (int s = 16; s > 0; s >>= 1) {
    if (c < s) red[c] += red[c + s];
    __syncthreads();
  }
  float inv = 1.0f / fmaxf(sqrtf(red[0]), 1e-12f);
  cXn[g * HD + c] = cx * inv;
  cC[g * HD + c] = cc;
}

// ---------------------------------------------------------------------------
// 3) cond encode GEMM via V_WMMA_F32_16X16X4_F32.
//    LDS B tile is K-pair swizzled: element (k, n) at ((k>>1)*32 + n)*2 + (k&1)
//    so each lane's B operand pair is one aligned ds_load_b64.
//    block = 8 waves; wave w -> rows [16w,16w+16); 32 columns per block
//    (two col tiles per wave -> two independent WMMA chains for ILP).
// ---------------------------------------------------------------------------
__global__ void __launch_bounds__(256) encode_kernel(
    const float* __restrict__ src, const float* __restrict__ W,
    const float* __restrict__ bias, float* __restrict__ feat) {
  __shared__ __align__(16) float Bs[CX * 32];
  const int tid = threadIdx.x;
  const int colBase = blockIdx.x * 32;          // 64000 % 32 == 0, no batch straddle
  const int b = colBase / NSP;
  const int spBase = colBase % NSP;
  for (int idx = tid; idx < (CX / 2) * 32; idx += 256) {   // 1024: (k-pair, col)
    int p = idx >> 5, nl = idx & 31;
    size_t g = (size_t)(b * CX + 2 * p) * NSP + spBase + nl;
    v2f v;
    v.x = src[g];
    v.y = src[g + NSP];
    *(v2f*)&Bs[(p * 32 + nl) * 2] = v;          // b64 store, conflict-free
  }
  __syncthreads();
  const int wave = tid >> 5, lane = tid & 31;
  const int lhalf = lane >> 4, l16 = lane & 15;
  const int m0 = wave * 16;
  v8f acc0 = {}, acc1 = {};
  for (int k0 = 0; k0 < CX; k0 += 4) {
    const int pk = (k0 >> 1) + lhalf;           // A/B reg j holds K = k0 + j + 2*(lane/16)
    v2f a  = *(const v2f*)&W[(m0 + l16) * CX + k0 + 2 * lhalf];
    v2f b0 = *(const v2f*)&Bs[(pk * 32 + l16) * 2];
    v2f b1 = *(const v2f*)&Bs[(pk * 32 + 16 + l16) * 2];
    acc0 = WMMA_F32_16x16x4(a, b0, acc0);
    acc1 = WMMA_F32_16x16x4(a, b1, acc1);
  }
#pragma unroll
  for (int r = 0; r < 8; ++r) {                 // D reg r: M = r + 8*(lane/16), N = lane%16
    int o = m0 + r + 8 * lhalf;
    float bo = bias[o];
    size_t base = (size_t)(b * EC + o) * NSP + spBase;
    feat[base + l16]      = acc0[r] + bo;
    feat[base + 16 + l16] = acc1[r] + bo;
  }
}

// ---------------------------------------------------------------------------
// 4) assign: per (bp, n) cosine sim vs 8 centers, sigmoid affine, hard argmax
//    (strict > keeps first index on ties, matching torch), accumulate num/den.
// ---------------------------------------------------------------------------
__global__ void __launch_bounds__(256) assign_kernel(
    const float* __restrict__ feat, const float* __restrict__ cXn,
    const float* __restrict__ alpha, const float* __restrict__ beta,
    float* __restrict__ simval, int* __restrict__ amax,
    float* __restrict__ numAcc, float* __restrict__ denAcc) {
  __shared__ float sC[M8 * HD];
  __shared__ float sAcc[M8 * (HD + 1)];
  const int tid = threadIdx.x;
  const int bp = blockIdx.y;
  const int b = bp >> 2, e = bp & 3;
  for (int i = tid; i < M8 * HD; i += 256) sC[i] = cXn[bp * M8 * HD + i];
  for (int i = tid; i < M8 * (HD + 1); i += 256) sAcc[i] = 0.0f;
  __syncthreads();

  const int n = blockIdx.x * 256 + tid;         // n-order (w,h,d)
  const int w = n / 1600, h = (n / 40) % 40, d = n % 40;
  const int sp = (h * S + w) * S + d;           // memory order (h,w,d)

  float cv[HD];
  float nrm2 = 0.0f;
#pragma unroll
  for (int c = 0; c < HD; ++c) {
    cv[c] = feat[(size_t)(b * EC + e * HD + c) * NSP + sp];
    nrm2 += cv[c] * cv[c];
  }
  const float inv = 1.0f / fmaxf(sqrtf(nrm2), 1e-12f);
  const float al = alpha[0], be = beta[0];

  float best = -1.0f;
  int bm = 0;
  for (int m = 0; m < M8; ++m) {
    float dot = 0.0f;
#pragma unroll
    for (int c = 0; c < HD; ++c) dot += sC[m * HD + c] * cv[c];
    float sv = 1.0f / (1.0f + __expf(-(be + al * dot * inv)));
    if (sv > best) { best = sv; bm = m; }
  }
  simval[(size_t)bp * NSP + n] = best;
  amax[(size_t)bp * NSP + n] = bm;

#pragma unroll
  for (int c = 0; c < HD; ++c) atomicAdd(&sAcc[bm * (HD + 1) + c], best * cv[c]);
  atomicAdd(&sAcc[bm * (HD + 1) + HD], best);
  __syncthreads();

  for (int i = tid; i < M8 * (HD + 1); i += 256) {
    int m = i / (HD + 1), j = i % (HD + 1);
    if (j < HD) atomicAdd(&numAcc[(bp * M8 + m) * HD + j], sAcc[i]);
    else        atomicAdd(&denAcc[bp * M8 + m], sAcc[i]);
  }
}

// ---------------------------------------------------------------------------
// 5) w = (num + cond_centers) / (den + 1)
// ---------------------------------------------------------------------------
__global__ void __launch_bounds__(256) wcalc_kernel(
    const float* __restrict__ num, const float* __restrict__ den,
    const float* __restrict__ cC, float* __restrict__ wbuf) {
  int id = blockIdx.x * 256 + threadIdx.x;
  if (id < NBP * M8 * HD) {
    int g = id / HD;                            // bp*8 + m
    wbuf[id] = (num[id] + cC[id]) / (den[g] + 1.0f);
  }
}

// ---------------------------------------------------------------------------
// 6) projection GEMM via WMMA; B built on the fly from (simval, argmax, w),
//    stored into the same K-pair-swizzled LDS layout. ch and ch+1 share the
//    same (simval, argmax) so each pair costs one gather.
//    grid = (4000, 2): x -> 32 columns, y -> 128-row half of M=256.
// ---------------------------------------------------------------------------
__global__ void __launch_bounds__(256) proj_kernel(
    const float* __restrict__ Wp, const float* __restrict__ bpj,
    const float* __restrict__ simval, const int* __restrict__ amax,
    const float* __restrict__ wbuf, float* __restrict__ out) {
  __shared__ __align__(16) float Bs[EC * 32];   // 16 KB, swizzled
  const int tid = threadIdx.x;
  const int colBase = blockIdx.x * 32;
  const int b = colBase / NSP;
  const int nBase = colBase % NSP;
  for (int idx = tid; idx < (EC / 2) * 32; idx += 256) {   // 2048: (ch-pair, col)
    int p = idx >> 5, nl = idx & 31;
    int ch = 2 * p;
    int e = ch >> 5, c = ch & 31;               // c, c+1 in same head
    int bpI = b * HEADS + e;
    size_t ni = (size_t)bpI * NSP + nBase + nl;
    float sv = simval[ni];
    int am = amax[ni];
    const float* wrow = &wbuf[(bpI * M8 + am) * HD];
    v2f v;
    v.x = sv * wrow[c];
    v.y = sv * wrow[c + 1];
    *(v2f*)&Bs[(p * 32 + nl) * 2] = v;
  }
  __syncthreads();
  const int wave = tid >> 5, lane = tid & 31;
  const int lhalf = lane >> 4, l16 = lane & 15;
  const int rowBase = blockIdx.y * 128 + wave * 16;
  v8f acc0 = {}, acc1 = {};
  for (int k0 = 0; k0 < EC; k0 += 4) {
    const int pk = (k0 >> 1) + lhalf;
    v2f a  = *(const v2f*)&Wp[(rowBase + l16) * EC + k0 + 2 * lhalf];
    v2f b0 = *(const v2f*)&Bs[(pk * 32 + l16) * 2];
    v2f b1 = *(const v2f*)&Bs[(pk * 32 + 16 + l16) * 2];
    acc0 = WMMA_F32_16x16x4(a, b0, acc0);
    acc1 = WMMA_F32_16x16x4(a, b1, acc1);
  }
#pragma unroll
  for (int r = 0; r < 8; ++r) {
    int o = rowBase + r + 8 * lhalf;
    float bo = bpj[o];
    size_t base = (size_t)(b * OUTC + o) * NSP + nBase;
    out[base + l16]      = acc0[r] + bo;
    out[base + 16 + l16] = acc1[r] + bo;
  }
}

// ---------------------------------------------------------------------------
// launch
// ---------------------------------------------------------------------------
extern "C" void kernel_launch(void* const* d_in, const int* in_sizes, int n_in,
                              void* d_out, int out_size, void* d_ws, size_t ws_size,
                              hipStream_t stream) {
  const float* x     = (const float*)d_in[0];
  const float* cond  = (const float*)d_in[1];
  const float* xw    = (const float*)d_in[2];
  const float* xb    = (const float*)d_in[3];
  const float* cw    = (const float*)d_in[4];
  const float* cb    = (const float*)d_in[5];
  const float* alpha = (const float*)d_in[6];
  const float* beta  = (const float*)d_in[7];
  const float* pw    = (const float*)d_in[8];
  const float* pb    = (const float*)d_in[9];
  float* out = (float*)d_out;

  // workspace layout (floats)
  float* ws = (float*)d_ws;
  size_t off = 0;
  float* condFeat = ws + off; off += (size_t)B0 * EC * NSP;      // 16,384,000
  float* simval   = ws + off; off += (size_t)NBP * NSP;          //    512,000
  int*   amax     = (int*)(ws + off); off += (size_t)NBP * NSP;  //    512,000
  float* xbar     = ws + off; off += B0 * CX * M8;               //      1,024
  float* cbar     = ws + off; off += B0 * CX * M8;               //      1,024
  float* cXn      = ws + off; off += NBP * M8 * HD;              //      2,048
  float* cC       = ws + off; off += NBP * M8 * HD;              //      2,048
  float* num      = ws + off; off += NBP * M8 * HD;              //      2,048
  float* den      = ws + off; off += NBP * M8;                   //         64
  float* wbuf     = ws + off; off += NBP * M8 * HD;              //      2,048

  zero_kernel<<<1, 256, 0, stream>>>(num, den);
  pool_kernel<<<dim3(CX, B0, 2), 256, 0, stream>>>(x, cond, xbar, cbar);
  centers_kernel<<<NBP * M8, 32, 0, stream>>>(xbar, cbar, xw, xb, cw, cb, cXn, cC);
  encode_kernel<<<(B0 * NSP) / 32, 256, 0, stream>>>(cond, cw, cb, condFeat);
  assign_kernel<<<dim3(NSP / 256, NBP), 256, 0, stream>>>(condFeat, cXn, alpha, beta,
                                                          simval, amax, num, den);
  wcalc_kernel<<<(NBP * M8 * HD + 255) / 256, 256, 0, stream>>>(num, den, cC, wbuf);
  proj_kernel<<<dim3((B0 * NSP) / 32, OUTC / 128), 256, 0, stream>>>(pw, pb, simval,
                                                                     amax, wbuf, out);
}